// EKTAttn_DSeqModel_61375082660571
// MI455X (gfx1250) — compile-verified
//
#include <hip/hip_runtime.h>
#include <hip/hip_bf16.h>

// Sizes from the reference
#define L_ 256
#define K_ 128
#define D_ 16
#define H_ 64
#define T_ 128
#define KE_ 64
#define DE_ 32
#define KD_ (K_ * D_)          // 2048
#define KDH_ (K_ * D_ * H_)    // 131072

typedef __attribute__((ext_vector_type(2))) float v2f;
typedef __attribute__((ext_vector_type(8))) float v8f;

// ---------------------------------------------------------------------------
// Phase A (1 block, 256 threads): attention softmax, coef, beta softmax, gxe
// ---------------------------------------------------------------------------
__global__ void __launch_bounds__(256)
phaseA_kernel(const float* __restrict__ text_v, const float* __restrict__ knowledge_v,
              const float* __restrict__ difficulty_v, const float* __restrict__ current_time,
              const float* __restrict__ result, const float* __restrict__ text_s,
              const float* __restrict__ difficulty_s, const float* __restrict__ knowledge_s,
              const float* __restrict__ times, const float* __restrict__ knowledge_memory,
              const float* __restrict__ difficulty_memory, const float* __restrict__ attention_weights,
              const float* __restrict__ time_decay_factor, const float* __restrict__ w_ih,
              float* __restrict__ coef_ws, float* __restrict__ beta_ws, float* __restrict__ gxe_ws)
{
    __shared__ float red[256];
    __shared__ float w3[3];
    __shared__ float ksim[K_];
    __shared__ float dsim[D_];
    const int t = threadIdx.x;

    if (t == 0) {
        float a0 = attention_weights[0], a1 = attention_weights[1], a2 = attention_weights[2];
        float m = fmaxf(a0, fmaxf(a1, a2));
        float e0 = expf(a0 - m), e1 = expf(a1 - m), e2 = expf(a2 - m);
        float s = e0 + e1 + e2;
        w3[0] = e0 / s; w3[1] = e1 / s; w3[2] = e2 / s;
    }
    __syncthreads();

    // per-l similarities (one l per thread)
    float ts = 0.f;
    {
        const float* r0 = text_s + t * T_;
        for (int i = 0; i < T_; ++i) ts += r0[i] * text_v[i];
    }
    float ds = 0.f;
    {
        const float* r0 = difficulty_s + t * DE_;
        for (int i = 0; i < DE_; ++i) ds += r0[i] * difficulty_v[i];
    }
    float ks = 0.f;
    {
        const float* r0 = knowledge_s + t * KE_;
        for (int i = 0; i < KE_; ++i) ks += r0[i] * knowledge_v[i];
    }
    float combined = w3[0] * ts + w3[1] * ds + w3[2] * ks;

    // softmax over L=256
    red[t] = combined; __syncthreads();
    for (int s = 128; s > 0; s >>= 1) { if (t < s) red[t] = fmaxf(red[t], red[t + s]); __syncthreads(); }
    float mx = red[0]; __syncthreads();
    float ex = expf(combined - mx);
    red[t] = ex; __syncthreads();
    for (int s = 128; s > 0; s >>= 1) { if (t < s) red[t] += red[t + s]; __syncthreads(); }
    float sum = red[0]; __syncthreads();
    float alpha = ex / sum;
    float decay = expf(-time_decay_factor[0] * (current_time[0] - times[t]));
    coef_ws[t] = alpha * decay;

    // ksim (128) and dsim (16)
    if (t < K_) {
        float a = 0.f; const float* km = knowledge_memory + t * KE_;
        for (int i = 0; i < KE_; ++i) a += km[i] * knowledge_v[i];
        ksim[t] = a;
    }
    if (t >= 128 && t < 128 + D_) {
        int d = t - 128;
        float a = 0.f; const float* dm = difficulty_memory + d * DE_;
        for (int i = 0; i < DE_; ++i) a += dm[i] * difficulty_v[i];
        dsim[d] = a;
    }
    __syncthreads();

    // beta = softmax over K*D = 2048 (8 entries per thread)
    float lg[8];
    float lmax = -INFINITY;
    for (int j = 0; j < 8; ++j) {
        int idx = t * 8 + j;
        lg[j] = ksim[idx >> 4] * dsim[idx & 15];
        lmax = fmaxf(lmax, lg[j]);
    }
    red[t] = lmax; __syncthreads();
    for (int s = 128; s > 0; s >>= 1) { if (t < s) red[t] = fmaxf(red[t], red[t + s]); __syncthreads(); }
    float bm = red[0]; __syncthreads();
    float lsum = 0.f;
    for (int j = 0; j < 8; ++j) { lg[j] = expf(lg[j] - bm); lsum += lg[j]; }
    red[t] = lsum; __syncthreads();
    for (int s = 128; s > 0; s >>= 1) { if (t < s) red[t] += red[t + s]; __syncthreads(); }
    float bs = red[0];
    for (int j = 0; j < 8; ++j) beta_ws[t * 8 + j] = lg[j] / bs;

    // gxe[j] = x_r . w_ih[j,:]  (rank-1 structure: x_r = [text_v,0] or [0,text_v])
    if (t < 3 * H_) {
        int off = (result[0] >= 0.5f) ? 0 : T_;
        const float* wr = w_ih + t * (2 * T_) + off;
        float a = 0.f;
        for (int i = 0; i < T_; ++i) a += text_v[i] * wr[i];
        gxe_ws[t] = a;
    }
}

// ---------------------------------------------------------------------------
// Phase B (128 blocks x 256 threads): H_att = sum_l coef[l] * hidden_s[l]
// Streams 134 MB — the bandwidth-bound hot loop (float4, coalesced).
// ---------------------------------------------------------------------------
__global__ void __launch_bounds__(256)
phaseB_kernel(const float* __restrict__ hidden_s, const float* __restrict__ coef_ws,
              float* __restrict__ Hatt_ws)
{
    __shared__ float coef[L_];
    const int t = threadIdx.x;
    coef[t] = coef_ws[t];
    __syncthreads();

    const int idx4 = blockIdx.x * 256 + t;              // 0 .. 32767 (float4 index)
    const float4* base = (const float4*)hidden_s + idx4; // stride per l = KDH_/4 float4
    float4 acc = make_float4(0.f, 0.f, 0.f, 0.f);
#pragma unroll 4
    for (int l = 0; l < L_; ++l) {
        float4 v = base[(size_t)l * (KDH_ / 4)];
        float c = coef[l];
        acc.x += c * v.x; acc.y += c * v.y; acc.z += c * v.z; acc.w += c * v.w;
    }
    ((float4*)Hatt_ws)[idx4] = acc;
}

// ---------------------------------------------------------------------------
// Phase C (1 block, 1024 threads): s_history, s_current, predict_score
// ---------------------------------------------------------------------------
__global__ void __launch_bounds__(1024)
phaseC_kernel(const float* __restrict__ hidden_last, const float* __restrict__ Hatt_ws,
              const float* __restrict__ beta_ws, const float* __restrict__ text_v,
              const float* __restrict__ difficulty_v, const float* __restrict__ knowledge_v,
              const float* __restrict__ score_w, const float* __restrict__ score_b,
              float* __restrict__ d_out)
{
    __shared__ float hist[16][H_];
    __shared__ float cur[16][H_];
    __shared__ float sh[H_], sc[H_];
    __shared__ float pr[T_ + DE_ + KE_ + 2 * H_];   // 352
    const int t = threadIdx.x;
    const int h = t & (H_ - 1);
    const int chunk = t >> 6;                        // 0..15, each covers 128 kd

    float ah = 0.f, ac = 0.f;
    for (int kd = chunk * 128; kd < chunk * 128 + 128; ++kd) {
        float b = beta_ws[kd];
        ah += b * Hatt_ws[kd * H_ + h];
        ac += b * hidden_last[kd * H_ + h];
    }
    hist[chunk][h] = ah; cur[chunk][h] = ac;
    __syncthreads();
    if (t < H_) {
        float s1 = 0.f, s2 = 0.f;
        for (int c = 0; c < 16; ++c) { s1 += hist[c][t]; s2 += cur[c][t]; }
        sh[t] = s1; sc[t] = s2;
    }
    __syncthreads();
    if (t < 352) {
        float v;
        if      (t < 128) v = text_v[t];
        else if (t < 160) v = difficulty_v[t - 128];
        else if (t < 224) v = knowledge_v[t - 160];
        else if (t < 288) v = sh[t - 224];
        else              v = sc[t - 288];
        pr[t] = score_w[t] * v;
    }
    __syncthreads();
    if (t == 0) {
        float s = score_b[0];
        for (int i = 0; i < 352; ++i) s += pr[i];
        d_out[0] = s;
    }
}

// ---------------------------------------------------------------------------
// Phase D (128 blocks x 128 threads): gh = hf @ w_hh^T via V_WMMA_F32_16X16X4_F32,
// fused with GRU gates -> h_new. One 16-row M-tile per block; 4 waves x 3 N-tiles.
// A 16x4 f32 layout: lanes 0-15 hold K=k0+{0,1}, lanes 16-31 hold K=k0+{2,3}.
// B 4x16 mirrored; C/D: vgpr r -> row r + (lane>=16)*8, col = lane&15.
// ---------------------------------------------------------------------------
__global__ void __launch_bounds__(128)
phaseD_kernel(const float* __restrict__ hidden_last, const float* __restrict__ w_hh,
              const float* __restrict__ beta_ws, const float* __restrict__ gxe_ws,
              const float* __restrict__ b_ih, const float* __restrict__ b_hh,
              float* __restrict__ d_out)
{
    __shared__ float ghtile[16 * 192];
    const int tid  = threadIdx.x;
    const int wave = tid >> 5;
    const int lane = tid & 31;
    const int ln   = lane & 15;
    const int hh   = lane >> 4;          // lane half selects K pair
    const int m0   = blockIdx.x * 16;

    const float* arow = hidden_last + (m0 + ln) * H_ + hh * 2;
    const int nt0 = wave * 3;            // waves cover n-tiles {0..2},{3..5},{6..8},{9..11}
    const float* brow0 = w_hh + ((nt0 + 0) * 16 + ln) * H_ + hh * 2;
    const float* brow1 = w_hh + ((nt0 + 1) * 16 + ln) * H_ + hh * 2;
    const float* brow2 = w_hh + ((nt0 + 2) * 16 + ln) * H_ + hh * 2;

    v8f acc0 = {}, acc1 = {}, acc2 = {};
#pragma unroll
    for (int k0 = 0; k0 < H_; k0 += 4) {
        v2f a  = *(const v2f*)(arow  + k0);
        v2f b0 = *(const v2f*)(brow0 + k0);
        v2f b1 = *(const v2f*)(brow1 + k0);
        v2f b2 = *(const v2f*)(brow2 + k0);
        acc0 = __builtin_amdgcn_wmma_f32_16x16x4_f32(false, a, false, b0, (short)0, acc0, false, false);
        acc1 = __builtin_amdgcn_wmma_f32_16x16x4_f32(false, a, false, b1, (short)0, acc1, false, false);
        acc2 = __builtin_amdgcn_wmma_f32_16x16x4_f32(false, a, false, b2, (short)0, acc2, false, false);
    }
#pragma unroll
    for (int r = 0; r < 8; ++r) {
        int row = r + hh * 8;
        ghtile[row * 192 + (nt0 + 0) * 16 + ln] = acc0[r];
        ghtile[row * 192 + (nt0 + 1) * 16 + ln] = acc1[r];
        ghtile[row * 192 + (nt0 + 2) * 16 + ln] = acc2[r];
    }
    __syncthreads();

    // GRU gates: 16 rows x 64 h = 1024 elems over 128 threads
    for (int e = tid; e < 16 * H_; e += 128) {
        int row = e >> 6;
        int h = e & (H_ - 1);
        int kd = m0 + row;
        float b = beta_ws[kd];
        float gxr = b * gxe_ws[h]           + b_ih[h];
        float gxz = b * gxe_ws[h + H_]      + b_ih[h + H_];
        float gxn = b * gxe_ws[h + 2 * H_]  + b_ih[h + 2 * H_];
        float ghr = ghtile[row * 192 + h]          + b_hh[h];
        float ghz = ghtile[row * 192 + h + H_]     + b_hh[h + H_];
        float ghn = ghtile[row * 192 + h + 2 * H_] + b_hh[h + 2 * H_];
        float rg = 1.f / (1.f + expf(-(gxr + ghr)));
        float zg = 1.f / (1.f + expf(-(gxz + ghz)));
        float ng = tanhf(gxn + rg * ghn);
        float hf = hidden_last[kd * H_ + h];
        d_out[1 + kd * H_ + h] = (1.f - zg) * ng + zg * hf;
    }
}

// ---------------------------------------------------------------------------
extern "C" void kernel_launch(void* const* d_in, const int* in_sizes, int n_in,
                              void* d_out, int out_size, void* d_ws, size_t ws_size,
                              hipStream_t stream) {
    const float* text_v            = (const float*)d_in[0];
    const float* knowledge_v       = (const float*)d_in[1];
    const float* difficulty_v      = (const float*)d_in[2];
    const float* current_time      = (const float*)d_in[3];
    const float* result            = (const float*)d_in[4];
    const float* text_s            = (const float*)d_in[5];
    const float* hidden_s          = (const float*)d_in[6];
    const float* difficulty_s      = (const float*)d_in[7];
    const float* knowledge_s       = (const float*)d_in[8];
    const float* times             = (const float*)d_in[9];
    const float* knowledge_memory  = (const float*)d_in[10];
    const float* difficulty_memory = (const float*)d_in[11];
    const float* attention_weights = (const float*)d_in[12];
    const float* time_decay_factor = (const float*)d_in[13];
    const float* w_ih              = (const float*)d_in[14];
    const float* w_hh              = (const float*)d_in[15];
    const float* b_ih              = (const float*)d_in[16];
    const float* b_hh              = (const float*)d_in[17];
    const float* score_w           = (const float*)d_in[18];
    const float* score_b           = (const float*)d_in[19];
    float* out = (float*)d_out;

    float* ws      = (float*)d_ws;
    float* coef_ws = ws;                 // 256
    float* beta_ws = ws + 256;           // 2048
    float* gxe_ws  = ws + 2304;          // 192
    float* Hatt_ws = ws + 2560;          // 131072 (16B-aligned: 2560*4 bytes)

    const float* hidden_last = hidden_s + (size_t)(L_ - 1) * KDH_;

    phaseA_kernel<<<1, 256, 0, stream>>>(text_v, knowledge_v, difficulty_v, current_time, result,
                                         text_s, difficulty_s, knowledge_s, times,
                                         knowledge_memory, difficulty_memory, attention_weights,
                                         time_decay_factor, w_ih, coef_ws, beta_ws, gxe_ws);
    phaseB_kernel<<<128, 256, 0, stream>>>(hidden_s, coef_ws, Hatt_ws);
    phaseC_kernel<<<1, 1024, 0, stream>>>(hidden_last, Hatt_ws, beta_ws, text_v, difficulty_v,
                                          knowledge_v, score_w, score_b, out);
    phaseD_kernel<<<128, 128, 0, stream>>>(hidden_last, w_hh, beta_ws, gxe_ws, b_ih, b_hh, out);
}